// FastDTWLoss_32701880992225
// MI455X (gfx1250) — compile-verified
//
#include <hip/hip_runtime.h>

// Exact DTW (N=4096) as an antidiagonal wavefront DP, resident on a single WGP.
//
//  - 2N-1 = 8191 strictly sequential diagonal steps -> latency-bound, so the
//    whole DP state lives in one 1024-thread workgroup (32 wave32 waves on one
//    WGP): per-step sync is a workgroup s_barrier, not a grid sync.
//  - Thread t owns rows 4t..4t+3; diagonals k-1 (d1) and k-2 (d2) are held in
//    registers. Only the last-row value crosses threads, exchanged through a
//    parity double-buffered, left-padded LDS array -> ONE barrier per diagonal
//    and an unconditional ds_load (slot 0 is a permanent BIG sentinel, so the
//    t==0 case needs no predication / exec masking).
//  - y[] (16 KB) is staged into LDS with the CDNA5 async global->LDS copy
//    (global_load_async_to_lds_b128 + s_wait_asynccnt).
//  - WMMA is not used: DTW is a (min,+)-semiring recurrence, which matrix
//    cores cannot express.

#define NSEQ  4096
#define TPB   1024
#define RPT   4                 // rows per thread; TPB*RPT == NSEQ
#define NDIAG (2 * NSEQ - 1)
#define BIGF  1e30f

typedef int v4i __attribute__((ext_vector_type(4)));
typedef v4i __attribute__((address_space(3))) as3_v4i;

__device__ __forceinline__ as3_v4i* to_lds_v4i(void* p) {
  // generic LDS pointer: low 32 bits are the LDS byte offset
  return (as3_v4i*)(unsigned int)(unsigned long long)p;
}

__launch_bounds__(TPB, 1)
__global__ void dtw4096_kernel(const float* __restrict__ xg,
                               const float* __restrict__ yg,
                               float* __restrict__ out) {
  __shared__ float ylds[NSEQ];        // whole target series, on-chip
  __shared__ float bnd[2][TPB + 1];   // +1: slot 0 is a BIG sentinel for t==0

  const int t  = threadIdx.x;
  const int r0 = t * RPT;             // first row owned by this thread

  // ---- stage y into LDS (CDNA5 async global->LDS copy, 16B per lane) ----
#if __has_builtin(__builtin_amdgcn_global_load_async_to_lds_b128) && \
    __has_builtin(__builtin_amdgcn_s_wait_asynccnt)
  __builtin_amdgcn_global_load_async_to_lds_b128(
      (v4i*)(yg + r0), to_lds_v4i(&ylds[r0]), /*offset=*/0, /*cpol=*/0);
  __builtin_amdgcn_s_wait_asynccnt(0);
#else
  #pragma unroll
  for (int q = 0; q < RPT; ++q) ylds[r0 + q] = yg[r0 + q];
#endif

  // ---- own rows of x into registers (one coalesced b128 load) ----
  const float4 xv = reinterpret_cast<const float4*>(xg)[t];
  float xr[RPT] = {xv.x, xv.y, xv.z, xv.w};

  float d1[RPT], d2[RPT];             // diagonals k-1 and k-2 (rows r0..r0+3)
  #pragma unroll
  for (int q = 0; q < RPT; ++q) { d1[q] = BIGF; d2[q] = BIGF; }
  float prev_left = BIGF;             // left neighbor's d2 boundary

  bnd[0][t + 1] = BIGF;
  bnd[1][t + 1] = BIGF;
  if (t == 0) { bnd[0][0] = BIGF; bnd[1][0] = BIGF; }  // permanent sentinel
  __syncthreads();

  // thread is provably all-BIG outside [k_on, k_off]
  const int k_on  = r0;
  const int k_off = r0 + (RPT - 1) + (NSEQ - 1);

  for (int k = 0; k < NDIAG; ++k) {
    const int wb = k & 1;             // slot written this step
    const int rb = wb ^ 1;            // slot written last step by neighbor
    const float left1 = bnd[rb][t];   // d1[r0-1]  (t==0 -> BIG sentinel)

    if (k >= k_on && k <= k_off) {
      float nd[RPT];
      #pragma unroll
      for (int q = 0; q < RPT; ++q) {
        const int i = r0 + q;
        const int j = k - i;                      // column on diagonal k
        const float up   = (q == 0) ? left1     : d1[q - 1];   // D[i-1, j]
        const float diag = (q == 0) ? prev_left : d2[q - 1];   // D[i-1, j-1]
        float m = fminf(d1[q], fminf(up, diag));               // + D[i, j-1]
        if (q == 0 && k == 0 && t == 0) m = 0.0f;              // D[0,0]=c[0,0]
        const bool valid = (unsigned)j < (unsigned)NSEQ;
        const float c    = fabsf(xr[q] - ylds[j & (NSEQ - 1)]); // masked idx
        nd[q] = valid ? (c + m) : BIGF;
      }
      #pragma unroll
      for (int q = 0; q < RPT; ++q) { d2[q] = d1[q]; d1[q] = nd[q]; }
      bnd[wb][t + 1] = d1[RPT - 1];
    } else {
      bnd[wb][t + 1] = BIGF;          // inactive rows are BIG on this diagonal
    }
    prev_left = left1;
    __syncthreads();                  // single barrier per diagonal
  }

  // D[N-1, N-1]: last diagonal, last row -> thread TPB-1, element RPT-1
  if (t == TPB - 1) out[0] = d1[RPT - 1];
}

extern "C" void kernel_launch(void* const* d_in, const int* in_sizes, int n_in,
                              void* d_out, int out_size, void* d_ws, size_t ws_size,
                              hipStream_t stream) {
  (void)in_sizes; (void)n_in; (void)out_size; (void)d_ws; (void)ws_size;
  const float* x = (const float*)d_in[0];   // "source"
  const float* y = (const float*)d_in[1];   // "target"
  float* out = (float*)d_out;               // scalar fp32 DTW distance
  dtw4096_kernel<<<dim3(1), dim3(TPB), 0, stream>>>(x, y, out);
}